// BoundaryLoss_15006615733788
// MI455X (gfx1250) — compile-verified
//
#include <hip/hip_runtime.h>
#include <hip/hip_bf16.h>

typedef __attribute__((ext_vector_type(2))) float v2f;
typedef __attribute__((ext_vector_type(8))) float v8f;

#define WDIM 256
#define HDIM 256
#define HW   65536
#define NC   4
#define NB   8
#define NS   32          // NB * NC slices
#define BIGF 1.0e10f     // matches reference INF

// ---------------------------------------------------------------- K0: init
__global__ void k_init(int* __restrict__ cnt, double* __restrict__ nd) {
    int t = threadIdx.x;
    if (t < NS) cnt[t] = 0;
    if (t == NS)     nd[0] = 0.0;   // numerator
    if (t == NS + 1) nd[1] = 0.0;   // denominator
}

// ---------------------------------------------------------------- K1: softmax over C (C=4)
__global__ void k_softmax(const float* __restrict__ pred, float* __restrict__ p) {
    int idx = blockIdx.x * 256 + threadIdx.x;      // 0 .. NB*HW-1
    int b = idx >> 16;
    int hw = idx & (HW - 1);
    const float* base = pred + (size_t)b * NC * HW + hw;
    float x0 = base[0], x1 = base[HW], x2 = base[2 * HW], x3 = base[3 * HW];
    float m = fmaxf(fmaxf(x0, x1), fmaxf(x2, x3));
    float e0 = expf(x0 - m), e1 = expf(x1 - m), e2 = expf(x2 - m), e3 = expf(x3 - m);
    float s = e0 + e1 + e2 + e3;
    float* ob = p + (size_t)b * NC * HW + hw;
    ob[0] = e0 / s; ob[HW] = e1 / s; ob[2 * HW] = e2 / s; ob[3 * HW] = e3 / s;
}

// ---------------------------------------------------------------- K2: |laplacian| via f32 WMMA
// lap = T*X + X*T,  T = tridiag(1,-2,1) (zero padding), per 256x256 slice.
// One wave computes one 16x16 output tile. Per tile: 12 v_wmma_f32_16x16x4_f32.
__device__ __forceinline__ float tval(int r, int c) {
    int d = r - c; d = d < 0 ? -d : d;
    return d == 0 ? -2.0f : (d == 1 ? 1.0f : 0.0f);
}

__global__ void k_boundary(const float* __restrict__ p, float* __restrict__ pb,
                           double* __restrict__ den) {
    int tid  = blockIdx.x * 256 + threadIdx.x;
    int wave = tid >> 5;
    int lane = threadIdx.x & 31;
    int s    = wave >> 8;            // slice 0..31
    int tile = wave & 255;
    int I = tile >> 4, J = tile & 15;
    int n = lane & 15, sel = lane >> 4;

    const float* X = p + (size_t)s * HW;
    int r0 = I * 16, c0 = J * 16;

    v8f acc = {0.f, 0.f, 0.f, 0.f, 0.f, 0.f, 0.f, 0.f};

    // B fragment: rows k0..k0+3 of the 16x16 tile at (rowBase,colBase)
    auto loadB = [&](int rowBase, int colBase, int k0) -> v2f {
        v2f bv;
        bv.x = X[(rowBase + k0 + 2 * sel) * WDIM + colBase + n];
        bv.y = X[(rowBase + k0 + 2 * sel + 1) * WDIM + colBase + n];
        return bv;
    };
    // A fragment: cols k0..k0+3 of the 16x16 tile at (rowBase,colBase)
    auto loadA = [&](int rowBase, int colBase, int k0) -> v2f {
        const float* q = X + (rowBase + n) * WDIM + colBase + k0 + 2 * sel;
        v2f av; av.x = q[0]; av.y = q[1];
        return av;
    };

    // ---- H direction: T * X ----
#pragma unroll
    for (int k0 = 0; k0 < 16; k0 += 4) {
        v2f a; a.x = tval(n, k0 + 2 * sel); a.y = tval(n, k0 + 2 * sel + 1);
        v2f b = loadB(r0, c0, k0);
        acc = __builtin_amdgcn_wmma_f32_16x16x4_f32(false, a, false, b, (short)0, acc, false, false);
    }
    if (I > 0) {   // T[I,I-1] = e0*e15^T : chunk k0=12, A[0][15]=1 -> a.y lane16
        v2f a; a.x = 0.f; a.y = (lane == 16) ? 1.f : 0.f;
        v2f b = loadB(r0 - 16, c0, 12);
        acc = __builtin_amdgcn_wmma_f32_16x16x4_f32(false, a, false, b, (short)0, acc, false, false);
    }
    if (I < 15) {  // T[I,I+1] = e15*e0^T : chunk k0=0, A[15][0]=1 -> a.x lane15
        v2f a; a.x = (lane == 15) ? 1.f : 0.f; a.y = 0.f;
        v2f b = loadB(r0 + 16, c0, 0);
        acc = __builtin_amdgcn_wmma_f32_16x16x4_f32(false, a, false, b, (short)0, acc, false, false);
    }
    // ---- W direction: X * T ----
#pragma unroll
    for (int k0 = 0; k0 < 16; k0 += 4) {
        v2f a = loadA(r0, c0, k0);
        v2f b; b.x = tval(k0 + 2 * sel, n); b.y = tval(k0 + 2 * sel + 1, n);
        acc = __builtin_amdgcn_wmma_f32_16x16x4_f32(false, a, false, b, (short)0, acc, false, false);
    }
    if (J > 0) {   // T[J-1,J][15][0]=1 : chunk k0=12 -> b.y lane16 (n==0)
        v2f a = loadA(r0, c0 - 16, 12);
        v2f b; b.x = 0.f; b.y = (lane == 16) ? 1.f : 0.f;
        acc = __builtin_amdgcn_wmma_f32_16x16x4_f32(false, a, false, b, (short)0, acc, false, false);
    }
    if (J < 15) {  // T[J+1,J][0][15]=1 : chunk k0=0 -> b.x lane15
        v2f a = loadA(r0, c0 + 16, 0);
        v2f b; b.x = (lane == 15) ? 1.f : 0.f; b.y = 0.f;
        acc = __builtin_amdgcn_wmma_f32_16x16x4_f32(false, a, false, b, (short)0, acc, false, false);
    }

    // |lap| -> pb, plus denominator partial sum
    float lsum = 0.f;
    float* O = pb + (size_t)s * HW;
#pragma unroll
    for (int r = 0; r < 8; ++r) {
        float v = fabsf(acc[r]);
        O[(r0 + r + 8 * sel) * WDIM + c0 + n] = v;
        lsum += v;
    }
    for (int off = 16; off; off >>= 1) lsum += __shfl_xor(lsum, off, 32);
    if (lane == 0) atomicAdd(den, (double)lsum);
}

// ---------------------------------------------------------------- K3: row-wise DT (binary seeds)
// g[b,c,h,x] = (dist along row to nearest target==c)^2, BIGF if none in row.
__global__ void k_rowdt(const int* __restrict__ target, float* __restrict__ g,
                        int* __restrict__ cnt) {
    int bh = blockIdx.x;           // 0..2047
    int b = bh >> 8, h = bh & 255;
    int x = threadIdx.x;
    __shared__ int cls[WDIM];
    __shared__ int scnt[NC];
    if (x < NC) scnt[x] = 0;
    __syncthreads();
    int cv = target[((size_t)b * HDIM + h) * WDIM + x] & 3;
    cls[x] = cv;
    atomicAdd(&scnt[cv], 1);
    __syncthreads();
#pragma unroll
    for (int c = 0; c < NC; ++c) {
        int r = -1;
        for (int d = 0; d < WDIM; ++d) {
            int lo = x - d, hi = x + d;
            bool hit = (lo >= 0 && cls[lo] == c) || (hi < WDIM && cls[hi] == c);
            if (hit) { r = d; break; }
        }
        float gv = (r < 0) ? BIGF : (float)(r * r);
        g[(((size_t)b * NC + c) * HDIM + h) * WDIM + x] = gv;
    }
    __syncthreads();
    if (x < NC) atomicAdd(&cnt[b * NC + x], scnt[x]);
}

// ---------------------------------------------------------------- K4: column DT + numerator
// One block per (slice, column). D[y] = min_j (y-j)^2 + g[j]; contrib = pb*sqrt(D).
__global__ void k_coldt(const float* __restrict__ g, const float* __restrict__ pb,
                        const int* __restrict__ cnt, double* __restrict__ num) {
    int blk = blockIdx.x;          // s*256 + x
    int s = blk >> 8;
    int x = blk & 255;
    int y = threadIdx.x;
    __shared__ float gcol[HDIM];
    __shared__ float red[256];
    gcol[y] = g[(size_t)s * HW + y * WDIM + x];
    __syncthreads();
    float m = 3.0e38f;
    float diff = (float)y;
#pragma unroll 4
    for (int j = 0; j < HDIM; ++j) {
        m = fminf(m, fmaf(diff, diff, gcol[j]));   // exact: diff^2 <= 65025 integer
        diff -= 1.0f;
    }
    float td = (cnt[s] > 0) ? sqrtf(m) : 0.0f;
    float contrib = pb[(size_t)s * HW + y * WDIM + x] * td;
    red[y] = contrib;
    __syncthreads();
    for (int off = 128; off; off >>= 1) {
        if (y < off) red[y] += red[y + off];
        __syncthreads();
    }
    if (y == 0) atomicAdd(num, (double)red[0]);
}

// ---------------------------------------------------------------- K5: finalize
__global__ void k_final(const double* __restrict__ nd, float* __restrict__ out) {
    out[0] = (float)(nd[0] / nd[1]);
}

// ---------------------------------------------------------------- launch
extern "C" void kernel_launch(void* const* d_in, const int* in_sizes, int n_in,
                              void* d_out, int out_size, void* d_ws, size_t ws_size,
                              hipStream_t stream) {
    const float* pred  = (const float*)d_in[0];
    const int* target  = (const int*)d_in[1];
    float* out = (float*)d_out;

    char* w = (char*)d_ws;
    float* p   = (float*)(w);                 // 8 MB softmax probs
    float* pb  = (float*)(w + 8388608);       // 8 MB |laplacian|
    float* g   = (float*)(w + 16777216);      // 8 MB row-pass squared distances
    int*   cnt = (int*)  (w + 25165824);      // 32 seed counts
    double* nd = (double*)(w + 25165952);     // [num, den]

    k_init    <<<1,    64, 0, stream>>>(cnt, nd);
    k_softmax <<<2048, 256, 0, stream>>>(pred, p);
    k_boundary<<<1024, 256, 0, stream>>>(p, pb, &nd[1]);
    k_rowdt   <<<2048, 256, 0, stream>>>(target, g, cnt);
    k_coldt   <<<8192, 256, 0, stream>>>(g, pb, cnt, &nd[0]);
    k_final   <<<1,    1,   0, stream>>>(nd, out);
}